// deformMP_28097676050585
// MI455X (gfx1250) — compile-verified
//
#include <hip/hip_runtime.h>
#include <stdint.h>
#include <math.h>

#define Bn   4
#define Cch  256
#define CI   64
#define Hh   128
#define Wwid 128
#define HW   16384
#define OFFC 18

typedef __attribute__((ext_vector_type(2))) float v2f;
typedef __attribute__((ext_vector_type(8))) float v8f;
typedef __attribute__((ext_vector_type(4))) unsigned int u32x4;
typedef __attribute__((ext_vector_type(8))) int i32x8;
typedef __attribute__((ext_vector_type(4))) int i32x4;

#if defined(__HIP_DEVICE_COMPILE__) && __has_builtin(__builtin_amdgcn_tensor_load_to_lds) && __has_builtin(__builtin_amdgcn_s_wait_tensorcnt)
#define HAVE_TDM 1
#else
#define HAVE_TDM 0
#endif

// Make an LDS buffer "escape" so the compiler must assume the TDM engine wrote
// it (the tensor_load_to_lds builtin only takes an integer LDS offset, so
// without this LLVM sees a never-stored shared array and folds its loads to
// undef -- observed in round 2's disasm as a loop-invariant WMMA B operand).
#define LDS_PUBLISH(p)                                   \
  do {                                                   \
    void* _e = (void*)(p);                               \
    asm volatile("" : "+v"(_e) :: "memory");             \
  } while (0)

// ---------------------------------------------------------------------------
// Tensor Data Mover: load a 2D fp32 tile (tile_w x tile_h rows) from a row-
// strided global tensor into LDS at byte offset lds_byte, padding each row of
// tile_w dwords with 1 extra dword (bank-conflict-free pitch = tile_w + 1).
// D# encoding per CDNA5 ISA 8.3/8.4.  pad_code: 3 -> 16-dword rows, 4 -> 32.
// ---------------------------------------------------------------------------
__device__ __forceinline__ void tdm_load_tile_f32(unsigned lds_byte,
                                                  const float* gsrc,
                                                  unsigned tile_w, unsigned tile_h,
                                                  unsigned tensor_w, unsigned tensor_h,
                                                  unsigned row_stride_elems,
                                                  int pad_code) {
#if HAVE_TDM
  unsigned long long ga = (unsigned long long)(uintptr_t)gsrc;
  u32x4 g0;
  g0[0] = 1u;                                                  // count=1, user D#
  g0[1] = lds_byte;                                            // lds_addr
  g0[2] = (unsigned)(ga & 0xFFFFFFFFu);                        // global_addr lo
  g0[3] = (unsigned)((ga >> 32) & 0x01FFFFFFu) | (2u << 30);   // addr hi | type=2
  i32x8 g1;
  g1[0] = (int)((2u << 16) |                   // data_size = 4 bytes
                (1u << 20) |                   // pad_enable
                ((unsigned)pad_code << 22) |   // pad_interval (2^(c+1) dwords)
                (0u << 25));                   // pad_amount = 1 dword
  g1[1] = (int)((tensor_w & 0xFFFFu) << 16);                       // dim0 lo16
  g1[2] = (int)((tensor_w >> 16) | ((tensor_h & 0xFFFFu) << 16));  // dim0 hi/dim1 lo
  g1[3] = (int)((tensor_h >> 16) | (tile_w << 16));                // dim1 hi/tile0
  g1[4] = (int)(tile_h & 0xFFFFu);                                 // tile1, tile2=0
  g1[5] = (int)row_stride_elems;                                   // dim0_stride lo32
  g1[6] = 0;
  g1[7] = 0;
  i32x4 z4 = {0, 0, 0, 0};
  i32x8 z8 = {0, 0, 0, 0, 0, 0, 0, 0};
  __builtin_amdgcn_tensor_load_to_lds(g0, g1, z4, z4, z8, 0);
  __builtin_amdgcn_s_wait_tensorcnt(0);
#else
  (void)lds_byte; (void)gsrc; (void)tile_w; (void)tile_h;
  (void)tensor_w; (void)tensor_h; (void)row_stride_elems; (void)pad_code;
#endif
}

// ---------------------------------------------------------------------------
// Kernel 1: y1[b,o,p] = sum_c w1[o,c] * x[b,c,p]    (M=64, K=256, N=HW)
// Block = 256 threads (8 waves): 4 M-tiles x 2 N-tiles of 16x16, 32 pixels/WG.
// ---------------------------------------------------------------------------
__global__ __launch_bounds__(256) void k1_conv1x1a(const float* __restrict__ x,
                                                   const float* __restrict__ w1,
                                                   float* __restrict__ y1) {
  __shared__ float lx[Cch * 33];          // [c][pixel] pitch 33 dwords
  int bid = blockIdx.x;
  int b   = bid >> 9;                     // 512 pixel-tiles per batch
  int p0  = (bid & 511) << 5;             // 32 pixels
  const float* src = x + (size_t)b * Cch * HW + p0;
#if HAVE_TDM
  unsigned lds_byte = (unsigned)(uintptr_t)(void*)&lx[0];  // LDS offset (low 32b)
  if (threadIdx.x < 32)
    tdm_load_tile_f32(lds_byte, src, 32u, (unsigned)Cch, (unsigned)HW,
                      (unsigned)Cch, (unsigned)HW, 4);
  LDS_PUBLISH(lx);
#else
  for (int i = threadIdx.x; i < Cch * 32; i += 256) {
    int c = i >> 5, p = i & 31;
    lx[c * 33 + p] = src[(size_t)c * HW + p];
  }
#endif
  __syncthreads();

  int wv = threadIdx.x >> 5, lane = threadIdx.x & 31;
  int l15 = lane & 15, hi = lane >> 4;
  int mt = wv & 3, nt = wv >> 2;
  int o = mt * 16 + l15;
  const float* wrow = w1 + o * Cch + 2 * hi;
  int pb = nt * 16 + l15;
  v8f acc = {0.f, 0.f, 0.f, 0.f, 0.f, 0.f, 0.f, 0.f};
#pragma unroll 4
  for (int k = 0; k < Cch; k += 4) {
    v2f a = *(const v2f*)(wrow + k);      // A[M=o][K=k+2hi .. +1]
    int kb = k + 2 * hi;
    v2f bv;
    bv.x = lx[kb * 33 + pb];              // B[K][N=pixel]
    bv.y = lx[(kb + 1) * 33 + pb];
    acc = __builtin_amdgcn_wmma_f32_16x16x4_f32(false, a, false, bv, (short)0,
                                                acc, false, false);
  }
  int pp = p0 + nt * 16 + l15;
  float* dst = y1 + (size_t)b * CI * HW + pp;
#pragma unroll
  for (int r = 0; r < 8; ++r) {
    int om = mt * 16 + r + 8 * hi;
    dst[(size_t)om * HW] = acc[r];
  }
}

// ---------------------------------------------------------------------------
// GroupNorm stats: one block per (b, group); writes mu, rsqrt(var+eps).
// ---------------------------------------------------------------------------
__global__ __launch_bounds__(256) void k_gnstats(const float* __restrict__ y,
                                                 float* __restrict__ st,
                                                 int cpg) {
  int bid = blockIdx.x;                       // b*32 + g
  const float* base = y + (size_t)bid * cpg * HW;
  int n = cpg * HW;
  float s = 0.f, s2 = 0.f;
  for (int i = threadIdx.x; i < n; i += 256) {
    float v = base[i];
    s += v; s2 += v * v;
  }
  __shared__ float rs[256], rq[256];
  rs[threadIdx.x] = s; rq[threadIdx.x] = s2;
  __syncthreads();
  for (int d = 128; d > 0; d >>= 1) {
    if (threadIdx.x < d) {
      rs[threadIdx.x] += rs[threadIdx.x + d];
      rq[threadIdx.x] += rq[threadIdx.x + d];
    }
    __syncthreads();
  }
  if (threadIdx.x == 0) {
    float inv = 1.f / (float)n;
    float mu = rs[0] * inv;
    float var = rq[0] * inv - mu * mu;
    st[bid * 2]     = mu;
    st[bid * 2 + 1] = rsqrtf(var + 1e-5f);
  }
}

// ---------------------------------------------------------------------------
// Normalize + ReLU for stage 1 (64 channels, 2 per group) -> xd
// ---------------------------------------------------------------------------
__global__ __launch_bounds__(256) void k3_norm_relu(const float* __restrict__ y1,
                                                    const float* __restrict__ st,
                                                    const float* __restrict__ gamma,
                                                    const float* __restrict__ beta,
                                                    float* __restrict__ xd) {
  int idx = blockIdx.x * 256 + threadIdx.x;
  int c = (idx >> 14) & 63;
  int b = idx >> 20;
  int sg = (b << 5) | (c >> 1);
  float mu = st[sg * 2], rsg = st[sg * 2 + 1];
  float v = (y1[idx] - mu) * rsg * gamma[c] + beta[c];
  xd[idx] = v > 0.f ? v : 0.f;
}

// ---------------------------------------------------------------------------
// 3x3 conv producing 18 offset channels (VALU; weights staged in LDS).
// ---------------------------------------------------------------------------
__global__ __launch_bounds__(256) void k4_conv3x3_off(const float* __restrict__ xd,
                                                      const float* __restrict__ w_off,
                                                      float* __restrict__ off) {
  __shared__ float lw[OFFC * CI * 9];
  for (int i = threadIdx.x; i < OFFC * CI * 9; i += 256) lw[i] = w_off[i];
  __syncthreads();
  int idx = blockIdx.x * 256 + threadIdx.x;
  int b = idx >> 14;
  int p = idx & (HW - 1);
  int h = p >> 7, w = p & 127;
  float acc[OFFC];
#pragma unroll
  for (int oc = 0; oc < OFFC; ++oc) acc[oc] = 0.f;
  const float* xb = xd + (size_t)b * CI * HW;
  for (int c = 0; c < CI; ++c) {
    const float* xc = xb + (size_t)c * HW;
#pragma unroll
    for (int t = 0; t < 9; ++t) {
      int yy = h + t / 3 - 1, xx = w + t % 3 - 1;
      float v = (yy >= 0 && yy < Hh && xx >= 0 && xx < Wwid) ? xc[(yy << 7) + xx]
                                                             : 0.f;
#pragma unroll
      for (int oc = 0; oc < OFFC; ++oc)
        acc[oc] = fmaf(lw[(oc * CI + c) * 9 + t], v, acc[oc]);
    }
  }
#pragma unroll
  for (int oc = 0; oc < OFFC; ++oc)
    off[((size_t)b * OFFC + oc) * HW + p] = acc[oc];
}

// ---------------------------------------------------------------------------
// Deformable conv: bilinear gather -> LDS im2col tile (576 x 16, pitch 17),
// then WMMA contraction (M=64, K=576), fused "+ xd" residual -> mid.
// Block = 128 threads (4 waves), 16 pixels per block.
// ---------------------------------------------------------------------------
__global__ __launch_bounds__(128) void k5_deform(const float* __restrict__ xd,
                                                 const float* __restrict__ off,
                                                 const float* __restrict__ w_def,
                                                 float* __restrict__ mid) {
  __shared__ float ls[576 * 17];
  int bid = blockIdx.x;
  int b = bid >> 10;                       // 1024 pixel-tiles per batch
  int p0 = (bid & 1023) << 4;              // 16 pixels (same image row)
  const float* xb = xd + (size_t)b * CI * HW;
  const float* ob = off + (size_t)b * OFFC * HW;

  for (int q = threadIdx.x; q < 144; q += 128) {   // (tap, pixel) pairs
    int tap = q >> 4;
    int p = q & 15;
    int pp = p0 + p;
    int h = pp >> 7, w = pp & 127;
    float dy = ob[(size_t)(tap * 2) * HW + pp];
    float dx = ob[(size_t)(tap * 2 + 1) * HW + pp];
    float py = (float)(h + tap / 3 - 1) + dy;
    float px = (float)(w + tap % 3 - 1) + dx;
    float y0f = floorf(py), x0f = floorf(px);
    float wy1 = py - y0f, wy0 = 1.f - wy1;
    float wx1 = px - x0f, wx0 = 1.f - wx1;
    int y0 = (int)y0f, x0 = (int)x0f;
    int y1i = y0 + 1, x1i = x0 + 1;
    int y0c = y0 < 0 ? 0 : (y0 > 127 ? 127 : y0);
    int y1c = y1i < 0 ? 0 : (y1i > 127 ? 127 : y1i);
    int x0c = x0 < 0 ? 0 : (x0 > 127 ? 127 : x0);
    int x1c = x1i < 0 ? 0 : (x1i > 127 ? 127 : x1i);
    bool vy0 = (y0 >= 0) && (y0 < 128);
    bool vy1 = (y1i >= 0) && (y1i < 128);
    bool vx0 = (x0 >= 0) && (x0 < 128);
    bool vx1 = (x1i >= 0) && (x1i < 128);
    float w00 = (vy0 && vx0) ? wy0 * wx0 : 0.f;
    float w01 = (vy0 && vx1) ? wy0 * wx1 : 0.f;
    float w10 = (vy1 && vx0) ? wy1 * wx0 : 0.f;
    float w11 = (vy1 && vx1) ? wy1 * wx1 : 0.f;
    int i00 = (y0c << 7) + x0c, i01 = (y0c << 7) + x1c;
    int i10 = (y1c << 7) + x0c, i11 = (y1c << 7) + x1c;
    for (int c = 0; c < CI; ++c) {
      const float* xc = xb + (size_t)c * HW;
      float v = w00 * xc[i00] + w01 * xc[i01] + w10 * xc[i10] + w11 * xc[i11];
      ls[(c * 9 + tap) * 17 + p] = v;      // K index = c*9 + tap
    }
  }
  __syncthreads();

  int wv = threadIdx.x >> 5;               // M-tile 0..3
  int lane = threadIdx.x & 31;
  int l15 = lane & 15, hi = lane >> 4;
  int o = wv * 16 + l15;
  const float* wrow = w_def + o * 576 + 2 * hi;
  v8f acc = {0.f, 0.f, 0.f, 0.f, 0.f, 0.f, 0.f, 0.f};
#pragma unroll 4
  for (int k = 0; k < 576; k += 4) {
    v2f a = *(const v2f*)(wrow + k);
    int kb = k + 2 * hi;
    v2f bv;
    bv.x = ls[kb * 17 + l15];
    bv.y = ls[(kb + 1) * 17 + l15];
    acc = __builtin_amdgcn_wmma_f32_16x16x4_f32(false, a, false, bv, (short)0,
                                                acc, false, false);
  }
  int pp = p0 + l15;
#pragma unroll
  for (int r = 0; r < 8; ++r) {
    int om = wv * 16 + r + 8 * hi;
    size_t gi = ((size_t)b * CI + om) * HW + pp;
    mid[gi] = acc[r] + xd[gi];
  }
}

// ---------------------------------------------------------------------------
// Kernel 6: y3[b,O,p] = sum_c w3[O,c] * mid[b,c,p]   (M=256, K=64)
// Block = 256 threads (8 waves), 16 pixels; each wave does 2 M-tiles.
// ---------------------------------------------------------------------------
__global__ __launch_bounds__(256) void k6_conv1x1b(const float* __restrict__ mid,
                                                   const float* __restrict__ w3,
                                                   float* __restrict__ y3) {
  __shared__ float lm[CI * 17];
  int bid = blockIdx.x;
  int b = bid >> 10;
  int p0 = (bid & 1023) << 4;
  const float* src = mid + (size_t)b * CI * HW + p0;
#if HAVE_TDM
  unsigned lds_byte = (unsigned)(uintptr_t)(void*)&lm[0];
  if (threadIdx.x < 32)
    tdm_load_tile_f32(lds_byte, src, 16u, (unsigned)CI, (unsigned)HW,
                      (unsigned)CI, (unsigned)HW, 3);
  LDS_PUBLISH(lm);
#else
  for (int i = threadIdx.x; i < CI * 16; i += 256) {
    int c = i >> 4, p = i & 15;
    lm[c * 17 + p] = src[(size_t)c * HW + p];
  }
#endif
  __syncthreads();
  int wv = threadIdx.x >> 5, lane = threadIdx.x & 31;
  int l15 = lane & 15, hi = lane >> 4;
#pragma unroll
  for (int half = 0; half < 2; ++half) {
    int mt = wv + half * 8;
    int o = mt * 16 + l15;
    const float* wrow = w3 + o * CI + 2 * hi;
    v8f acc = {0.f, 0.f, 0.f, 0.f, 0.f, 0.f, 0.f, 0.f};
#pragma unroll
    for (int k = 0; k < CI; k += 4) {
      v2f a = *(const v2f*)(wrow + k);
      int kb = k + 2 * hi;
      v2f bv;
      bv.x = lm[kb * 17 + l15];
      bv.y = lm[(kb + 1) * 17 + l15];
      acc = __builtin_amdgcn_wmma_f32_16x16x4_f32(false, a, false, bv, (short)0,
                                                  acc, false, false);
    }
    int pp = p0 + l15;
#pragma unroll
    for (int r = 0; r < 8; ++r) {
      int om = mt * 16 + r + 8 * hi;
      y3[((size_t)b * Cch + om) * HW + pp] = acc[r];
    }
  }
}

// ---------------------------------------------------------------------------
// Final: out = relu(GN(y3)*g3+b3) + x
// ---------------------------------------------------------------------------
__global__ __launch_bounds__(256) void k8_final(const float* __restrict__ y3,
                                                const float* __restrict__ st,
                                                const float* __restrict__ g3,
                                                const float* __restrict__ b3,
                                                const float* __restrict__ x,
                                                float* __restrict__ out) {
  int idx = blockIdx.x * 256 + threadIdx.x;
  int c = (idx >> 14) & 255;
  int b = idx >> 22;
  int sg = (b << 5) | (c >> 3);
  float mu = st[sg * 2], rsg = st[sg * 2 + 1];
  float v = (y3[idx] - mu) * rsg * g3[c] + b3[c];
  v = v > 0.f ? v : 0.f;
  out[idx] = v + x[idx];
}

// ---------------------------------------------------------------------------
extern "C" void kernel_launch(void* const* d_in, const int* in_sizes, int n_in,
                              void* d_out, int out_size, void* d_ws, size_t ws_size,
                              hipStream_t stream) {
  (void)in_sizes; (void)n_in; (void)out_size; (void)ws_size;
  const float* x    = (const float*)d_in[0];
  const float* w1   = (const float*)d_in[1];
  const float* g1   = (const float*)d_in[2];
  const float* b1   = (const float*)d_in[3];
  const float* woff = (const float*)d_in[4];
  const float* wdef = (const float*)d_in[5];
  const float* w3   = (const float*)d_in[6];
  const float* g3   = (const float*)d_in[7];
  const float* b3   = (const float*)d_in[8];
  float* out = (float*)d_out;

  float* ws  = (float*)d_ws;
  float* y1  = ws;                                   // B*CI*HW
  float* xd  = y1 + (size_t)Bn * CI * HW;            // B*CI*HW
  float* off = xd + (size_t)Bn * CI * HW;            // B*18*HW
  float* mid = off + (size_t)Bn * OFFC * HW;         // B*CI*HW
  float* y3  = mid + (size_t)Bn * CI * HW;           // B*Cch*HW
  float* st1 = y3 + (size_t)Bn * Cch * HW;           // 256
  float* st3 = st1 + 256;                            // 256

  k1_conv1x1a<<<Bn * 512, 256, 0, stream>>>(x, w1, y1);
  k_gnstats<<<Bn * 32, 256, 0, stream>>>(y1, st1, 2);
  k3_norm_relu<<<(Bn * CI * HW) / 256, 256, 0, stream>>>(y1, st1, g1, b1, xd);
  k4_conv3x3_off<<<(Bn * HW) / 256, 256, 0, stream>>>(xd, woff, off);
  k5_deform<<<Bn * 1024, 128, 0, stream>>>(xd, off, wdef, mid);
  k6_conv1x1b<<<Bn * 1024, 256, 0, stream>>>(mid, w3, y3);
  k_gnstats<<<Bn * 32, 256, 0, stream>>>(y3, st3, 8);
  k8_final<<<(Bn * Cch * HW) / 256, 256, 0, stream>>>(y3, st3, g3, b3, x, out);
}